// RGCNCell_7017976561679
// MI455X (gfx1250) — compile-verified
//
#include <hip/hip_runtime.h>

typedef float v2f __attribute__((ext_vector_type(2)));
typedef float v8f __attribute__((ext_vector_type(8)));

#define NN 50000
#define EE 800000
#define DD 200
#define MT_TOTAL (NN / 16)          // 3125 row tiles
#define NTILES ((DD + 15) / 16)     // 13 col tiles (last partial: 200 = 12*16+8)
#define RRELU_SLOPE ((1.0f/8.0f + 1.0f/3.0f) * 0.5f)

// h0[n,d] = emb[node_id[n], d]   (float2 per thread)
__global__ void rgcn_gather_h0(const float* __restrict__ emb,
                               const int* __restrict__ node_id,
                               float* __restrict__ h0) {
    long t = (long)blockIdx.x * blockDim.x + threadIdx.x;
    if (t >= (long)NN * (DD / 2)) return;
    int row = (int)(t / (DD / 2));
    int d2  = (int)(t - (long)row * (DD / 2)) * 2;
    const v2f v = *(const v2f*)(emb + (long)node_id[row] * DD + d2);
    *(v2f*)(h0 + (long)row * DD + d2) = v;
}

// mask[dst[e]] = 1  (idempotent plain store; race-free)
__global__ void rgcn_indeg_mask(const int* __restrict__ dst,
                                float* __restrict__ mask) {
    int e = blockIdx.x * blockDim.x + threadIdx.x;
    if (e < EE) mask[dst[e]] = 1.0f;
}

// pre[dst[e], d] += h[src[e], d] + rel[etype[e], d]   (float2 per thread)
__global__ void rgcn_scatter(const float* __restrict__ h,
                             const float* __restrict__ rel,
                             const int* __restrict__ src,
                             const int* __restrict__ dstv,
                             const int* __restrict__ etype,
                             float* __restrict__ pre) {
    long t = (long)blockIdx.x * blockDim.x + threadIdx.x;
    if (t >= (long)EE * (DD / 2)) return;
    int e  = (int)(t / (DD / 2));
    int d2 = (int)(t - (long)e * (DD / 2)) * 2;
    const v2f hv = *(const v2f*)(h   + (long)src[e]   * DD + d2);
    const v2f rv = *(const v2f*)(rel + (long)etype[e] * DD + d2);
    float* p = pre + (long)dstv[e] * DD + d2;
    unsafeAtomicAdd(p,     hv.x + rv.x);
    unsafeAtomicAdd(p + 1, hv.y + rv.y);
}

// Fused per-layer kernel: one wave per 16x16 output tile, 8 waves/block.
// B tiles (3 x 200x16 weight slices) staged once per block into LDS as
// float2 pairs {W[2k][c], W[2k+1][c]}.
//   agg = pre @ Wn ; la = h @ Wl ; le = h @ We   (v_wmma_f32_16x16x4_f32)
//   out = rrelu(agg*norm + (mask ? la : le))
__global__ void __launch_bounds__(256)
rgcn_layer(const float* __restrict__ pre,
           const float* __restrict__ h,
           const float* __restrict__ Wn,
           const float* __restrict__ Wl,
           const float* __restrict__ We,
           const float* __restrict__ norm,
           const float* __restrict__ mask,
           float* __restrict__ out) {
    __shared__ v2f sB[3][(DD / 2) * 16];   // [mat][(k/2)*16 + col] ; 37.5 KB

    const int lane  = threadIdx.x & 31;
    const int wave  = threadIdx.x >> 5;
    const int nbase = blockIdx.y * 16;

    // ---- stage B tiles (guard partial column tile here, outside hot loop) ----
    const float* const Ws[3] = { Wn, Wl, We };
    for (int i = threadIdx.x; i < 3 * (DD / 2) * 16; i += 256) {
        const int m  = i / ((DD / 2) * 16);
        const int r  = i - m * ((DD / 2) * 16);
        const int kp = r >> 4;          // K pair index
        const int c  = r & 15;
        const int col = nbase + c;
        v2f v = { 0.f, 0.f };
        if (col < DD) {
            const float* W = Ws[m];
            v.x = W[(long)(2 * kp)     * DD + col];
            v.y = W[(long)(2 * kp + 1) * DD + col];
        }
        sB[m][r] = v;
    }
    __syncthreads();

    // ---- per-wave 16x16 tile ----
    int mtile = blockIdx.x * 8 + wave;
    const bool mvalid = (mtile < MT_TOTAL);
    if (!mvalid) mtile = MT_TOTAL - 1;   // clamp: compute garbage, suppress store
    const int mbase = mtile * 16;

    // A operand (16x4 f32): lanes 0-15 K={0,1}, lanes 16-31 K={2,3}
    const int arow = mbase + (lane & 15);
    const int akk  = (lane >> 4) * 2;
    const int c    = lane & 15;

    v8f acc_n = {0.f,0.f,0.f,0.f,0.f,0.f,0.f,0.f};
    v8f acc_l = acc_n;
    v8f acc_e = acc_n;

    const float* pA = pre + (long)arow * DD + akk;
    const float* hA = h   + (long)arow * DD + akk;
    const v2f* B0 = &sB[0][c];
    const v2f* B1 = &sB[1][c];
    const v2f* B2 = &sB[2][c];
    const int kpofs = akk >> 1;          // 0 or 1

    #pragma unroll 5
    for (int k = 0; k < DD; k += 4) {
        const v2f a_pre = *(const v2f*)(pA + k);   // 8B aligned: all offsets even
        const v2f a_h   = *(const v2f*)(hA + k);
        const int kp = (k >> 1) + kpofs;
        const v2f b_n = B0[kp * 16];
        const v2f b_l = B1[kp * 16];
        const v2f b_e = B2[kp * 16];

        acc_n = __builtin_amdgcn_wmma_f32_16x16x4_f32(false, a_pre, false, b_n,
                                                      (short)0, acc_n, false, false);
        acc_l = __builtin_amdgcn_wmma_f32_16x16x4_f32(false, a_h,   false, b_l,
                                                      (short)0, acc_l, false, false);
        acc_e = __builtin_amdgcn_wmma_f32_16x16x4_f32(false, a_h,   false, b_e,
                                                      (short)0, acc_e, false, false);
    }

    // ---- epilogue ----
    // C/D layout: VGPR r -> M = r (lanes 0-15) or r+8 (lanes 16-31), N = lane&15
    if (mvalid) {                        // wave-uniform
        const int ocol = nbase + c;
        if (ocol < DD) {                 // lane guard (partial last col tile)
            const int half = (lane >> 4) * 8;
            #pragma unroll
            for (int r = 0; r < 8; ++r) {
                const int row = mbase + half + r;
                const float nm = norm[row];
                const float lm = (mask[row] != 0.0f) ? acc_l[r] : acc_e[r];
                float v = acc_n[r] * nm + lm;
                v = (v >= 0.0f) ? v : v * RRELU_SLOPE;
                out[(long)row * DD + ocol] = v;
            }
        }
    }
}

extern "C" void kernel_launch(void* const* d_in, const int* in_sizes, int n_in,
                              void* d_out, int out_size, void* d_ws, size_t ws_size,
                              hipStream_t stream) {
    const float* emb  = (const float*)d_in[0];   // [N,D]
    const float* rel  = (const float*)d_in[1];   // [R,D]
    const float* Wn   = (const float*)d_in[2];   // [L,D,D]
    const float* Wl   = (const float*)d_in[3];   // [L,D,D]
    const float* We   = (const float*)d_in[4];   // [L,D,D]
    const float* norm = (const float*)d_in[5];   // [N,1]
    const int*   src  = (const int*)d_in[6];
    const int*   dst  = (const int*)d_in[7];
    const int*   et   = (const int*)d_in[8];
    const int*   nid  = (const int*)d_in[9];
    float*       out  = (float*)d_out;           // [N,D]

    const size_t nd = (size_t)NN * DD;
    float* h0   = (float*)d_ws;
    float* h1   = h0  + nd;
    float* pre  = h1  + nd;
    float* mask = pre + nd;

    const int tpb = 256;

    hipMemsetAsync(mask, 0, NN * sizeof(float), stream);
    rgcn_gather_h0<<<(unsigned)((nd / 2 + tpb - 1) / tpb), tpb, 0, stream>>>(emb, nid, h0);
    rgcn_indeg_mask<<<(EE + tpb - 1) / tpb, tpb, 0, stream>>>(dst, mask);

    const float* hcur = h0;
    float* houts[2] = { h1, out };
    const size_t wstride = (size_t)DD * DD;

    for (int l = 0; l < 2; ++l) {
        hipMemsetAsync(pre, 0, nd * sizeof(float), stream);
        rgcn_scatter<<<(unsigned)(((size_t)EE * (DD / 2) + tpb - 1) / tpb), tpb, 0, stream>>>(
            hcur, rel, src, dst, et, pre);

        dim3 grid((MT_TOTAL + 7) / 8, NTILES);   // 391 x 13
        rgcn_layer<<<grid, 256, 0, stream>>>(pre, hcur,
            Wn + (size_t)l * wstride, Wl + (size_t)l * wstride, We + (size_t)l * wstride,
            norm, mask, houts[l]);

        hcur = houts[l];
    }
}